// GraphNeuralNetwork_2018634629265
// MI455X (gfx1250) — compile-verified
//
#include <hip/hip_runtime.h>
#include <stdint.h>

typedef __attribute__((ext_vector_type(2))) float v2f;
typedef __attribute__((ext_vector_type(8))) float v8f;

// ---------------------------------------------------------------- degree ----
__global__ void init_deg_kernel(float* deg, int n) {
  int i = blockIdx.x * blockDim.x + threadIdx.x;
  if (i < n) deg[i] = 1.0f;                       // self-loop contributes 1
}

__global__ void deg_edges_kernel(const int64_t* __restrict__ dst, float* deg, int e) {
  int i = blockIdx.x * blockDim.x + threadIdx.x;
  if (i < e) atomicAdd(&deg[(int)dst[i]], 1.0f);
}

__global__ void finalize_dinv_kernel(float* deg, int n) {
  int i = blockIdx.x * blockDim.x + threadIdx.x;
  if (i < n) deg[i] = rsqrtf(deg[i]);             // deg >= 1 always (self-loop)
}

// ------------------------------------------------------------- WMMA GEMM ----
// out[M,N] = A[M,128] @ W[128,N] (+bias, +relu).  N compile-time (128 or 64).
// blockDim=(32, N/16); block covers a 16-row x N tile. The 16x128 A tile is
// staged to LDS once (shared by all waves); rows padded to 132 floats so the
// ds_load_b64 A-fragment reads are bank-conflict free. B/out use immediate
// offsets (compile-time N). One wave = one 16x16 tile = 32 chained
// V_WMMA_F32_16X16X4_F32.
template <int N>
__global__ void __launch_bounds__(2 * N)
gemm_wmma_kernel(const float* __restrict__ A,
                 const float* __restrict__ W,
                 const float* __restrict__ bias,
                 float* __restrict__ out, int doRelu)
{
  __shared__ float As[16][132];

  const int lane = threadIdx.x;                   // 0..31
  const int tid  = threadIdx.y * 32 + lane;       // 0..2N-1
  const int row0 = blockIdx.x * 16;

  // ---- stage A tile (16 x 128 = 512 float4 chunks) into LDS ----
  {
    const float4* Ab = (const float4*)(A + (size_t)row0 * 128);
#pragma unroll
    for (int c = tid; c < 512; c += 2 * N) {
      int r = c >> 5;                             // row 0..15
      int f = (c & 31) << 2;                      // feature 0,4,...,124
      float4 v = Ab[(size_t)r * 32 + (c & 31)];
      As[r][f + 0] = v.x; As[r][f + 1] = v.y;
      As[r][f + 2] = v.z; As[r][f + 3] = v.w;
    }
  }
  __syncthreads();

  const int col0  = threadIdx.y * 16;
  const int mrow  = lane & 15;
  const int khalf = (lane >> 4) << 1;             // 0 or 2

  v8f c = {};
  const float* arow = &As[mrow][khalf];
  const float* bcol = W + (size_t)khalf * N + (col0 + mrow);

#pragma unroll
  for (int k = 0; k < 128; k += 4) {
    v2f a = *(const v2f*)(arow + k);              // LDS, 8B aligned
    v2f b;
    b.x = bcol[k * N];                            // immediate offsets
    b.y = bcol[(k + 1) * N];
    c = __builtin_amdgcn_wmma_f32_16x16x4_f32(
        /*neg_a=*/false, a, /*neg_b=*/false, b,
        /*c_mod=*/(short)0, c, /*reuse_a=*/false, /*reuse_b=*/false);
  }

  const int   ncol  = col0 + mrow;
  const int   rbase = row0 + ((lane >> 4) << 3);
  const float bv    = bias ? bias[ncol] : 0.0f;
  float* orow = out + (size_t)rbase * N + ncol;
#pragma unroll
  for (int r = 0; r < 8; ++r) {
    float v = c[r] + bv;
    if (doRelu) v = fmaxf(v, 0.0f);
    orow[r * N] = v;                              // immediate offsets
  }
}

// ------------------------------------------------------ GCN aggregation -----
// acc[i][:] = xw[i][:] * dinv[i]^2   (self-loop term, also initializes acc)
__global__ void selfloop_init_kernel(const float4* __restrict__ xw,
                                     const float* __restrict__ dinv,
                                     float4* __restrict__ acc, int nchunks)
{
  int idx = blockIdx.x * blockDim.x + threadIdx.x;
  if (idx >= nchunks) return;
  int node = idx >> 5;                            // 32 float4 chunks per node
  float d = dinv[node];
  float s = d * d;
  float4 v = xw[idx];
  v.x *= s; v.y *= s; v.z *= s; v.w *= s;
  acc[idx] = v;
}

// One wave per edge (blockDim=(32,8): edge id is wave-uniform so src/dst
// loads scalarize). Lanes cover 128 feats as float4: gather is a coalesced
// 512B wave load that hits L2 (feature matrix fits in 192MB L2); scatter is
// 4x global_atomic_add_f32 per lane.
__global__ void __launch_bounds__(256)
edge_scatter_kernel(const int64_t* __restrict__ src,
                    const int64_t* __restrict__ dst,
                    const float* __restrict__ dinv,
                    const float4* __restrict__ xw,
                    float* __restrict__ acc, int e)
{
  int wave = blockIdx.x * 8 + threadIdx.y;
  int lane = threadIdx.x;
  if (wave >= e) return;
  int s = (int)src[wave];
  int d = (int)dst[wave];
  float nrm = dinv[s] * dinv[d];
  float4 m = xw[(size_t)s * 32 + lane];
  float* o = acc + (size_t)d * 128 + lane * 4;
  atomicAdd(o + 0, m.x * nrm);
  atomicAdd(o + 1, m.y * nrm);
  atomicAdd(o + 2, m.z * nrm);
  atomicAdd(o + 3, m.w * nrm);
}

__global__ void bias_relu_kernel(float* acc, const float* __restrict__ bias, int total) {
  int idx = blockIdx.x * blockDim.x + threadIdx.x;
  if (idx >= total) return;
  acc[idx] = fmaxf(acc[idx] + bias[idx & 127], 0.0f);
}

// ---------------------------------------------------------------- pooling ---
__global__ void zero_kernel(float* p, int n) {
  int i = blockIdx.x * blockDim.x + threadIdx.x;
  if (i < n) p[i] = 0.0f;
}

__global__ void __launch_bounds__(256)
pool_accum_kernel(const float4* __restrict__ h,
                  const int64_t* __restrict__ batch,
                  float* __restrict__ gsum, float* __restrict__ gcnt, int n)
{
  int wave = blockIdx.x * 8 + threadIdx.y;        // node id, wave-uniform
  int lane = threadIdx.x;
  if (wave >= n) return;
  int gi = (int)batch[wave];
  float4 v = h[(size_t)wave * 32 + lane];
  float* o = gsum + (size_t)gi * 128 + lane * 4;
  atomicAdd(o + 0, v.x);
  atomicAdd(o + 1, v.y);
  atomicAdd(o + 2, v.z);
  atomicAdd(o + 3, v.w);
  if (lane == 0) atomicAdd(&gcnt[gi], 1.0f);
}

__global__ void pool_final_kernel(const float* __restrict__ gsum,
                                  const float* __restrict__ gcnt,
                                  float* __restrict__ g, int total)
{
  int idx = blockIdx.x * blockDim.x + threadIdx.x;
  if (idx >= total) return;
  g[idx] = gsum[idx] / fmaxf(gcnt[idx >> 7], 1.0f);
}

// ----------------------------------------------------------------- driver ---
extern "C" void kernel_launch(void* const* d_in, const int* in_sizes, int n_in,
                              void* d_out, int out_size, void* d_ws, size_t ws_size,
                              hipStream_t stream)
{
  const float*   x     = (const float*)d_in[0];
  const int64_t* ei    = (const int64_t*)d_in[1];
  const int64_t* batch = (const int64_t*)d_in[2];
  const float*   W1    = (const float*)d_in[3];
  const float*   b1    = (const float*)d_in[4];
  const float*   W2    = (const float*)d_in[5];
  const float*   b2    = (const float*)d_in[6];
  const float*   fw1   = (const float*)d_in[7];
  const float*   fb1   = (const float*)d_in[8];
  const float*   fw2   = (const float*)d_in[9];
  const float*   fb2   = (const float*)d_in[10];
  float*         out   = (float*)d_out;

  const int N    = in_sizes[0] / 128;   // 50000 nodes
  const int E    = in_sizes[1] / 2;     // 800000 edges
  const int dOut = in_sizes[10];        // 64
  const int G    = out_size / dOut;     // 64 graphs
  const int64_t* src = ei;
  const int64_t* dst = ei + E;

  // workspace carve-up (floats)
  float* ws = (float*)d_ws;
  size_t off = 0;
  float* dinv = ws + off; off += ((size_t)N + 63) & ~(size_t)63;
  float* buf0 = ws + off; off += (size_t)N * 128;   // GEMM output (xw)
  float* buf1 = ws + off; off += (size_t)N * 128;   // aggregation acc / h
  float* gsum = ws + off; off += (size_t)G * 128;
  float* gcnt = ws + off; off += 64;
  float* gbuf = ws + off; off += (size_t)G * 128;
  float* fc1  = ws + off; off += (size_t)G * 128;
  (void)ws_size; (void)n_in;

  const int T = 256;
  const int nodeBlocks  = (N + T - 1) / T;
  const int edgeBlocks  = (E + T - 1) / T;
  const int waveBlksE   = (E + 7) / 8;            // 8 waves / 256-thr block
  const int waveBlksN   = (N + 7) / 8;
  const int chunkBlocks = ((N * 32) + T - 1) / T; // float4 chunks
  const int featBlocks  = ((N * 128) + T - 1) / T;

  // degrees -> dinv
  init_deg_kernel<<<nodeBlocks, T, 0, stream>>>(dinv, N);
  deg_edges_kernel<<<edgeBlocks, T, 0, stream>>>(dst, dinv, E);
  finalize_dinv_kernel<<<nodeBlocks, T, 0, stream>>>(dinv, N);

  // layer 1: xw = x @ W1 ; aggregate ; bias+relu
  gemm_wmma_kernel<128><<<dim3(N / 16), dim3(32, 8), 0, stream>>>(x, W1, nullptr, buf0, 0);
  selfloop_init_kernel<<<chunkBlocks, T, 0, stream>>>((const float4*)buf0, dinv, (float4*)buf1, N * 32);
  edge_scatter_kernel<<<waveBlksE, dim3(32, 8), 0, stream>>>(src, dst, dinv, (const float4*)buf0, buf1, E);
  bias_relu_kernel<<<featBlocks, T, 0, stream>>>(buf1, b1, N * 128);

  // layer 2: hw = h @ W2 ; aggregate ; bias+relu   (buf1 reused after GEMM)
  gemm_wmma_kernel<128><<<dim3(N / 16), dim3(32, 8), 0, stream>>>(buf1, W2, nullptr, buf0, 0);
  selfloop_init_kernel<<<chunkBlocks, T, 0, stream>>>((const float4*)buf0, dinv, (float4*)buf1, N * 32);
  edge_scatter_kernel<<<waveBlksE, dim3(32, 8), 0, stream>>>(src, dst, dinv, (const float4*)buf0, buf1, E);
  bias_relu_kernel<<<featBlocks, T, 0, stream>>>(buf1, b2, N * 128);

  // global mean pool
  zero_kernel<<<((G * 128 + 64) + T - 1) / T, T, 0, stream>>>(gsum, G * 128 + 64); // gsum+gcnt contiguous
  pool_accum_kernel<<<waveBlksN, dim3(32, 8), 0, stream>>>((const float4*)buf1, batch, gsum, gcnt, N);
  pool_final_kernel<<<((G * 128) + T - 1) / T, T, 0, stream>>>(gsum, gcnt, gbuf, G * 128);

  // MLP head (fused bias, relu on first)
  gemm_wmma_kernel<128><<<dim3(G / 16), dim3(32, 8), 0, stream>>>(gbuf, fw1, fb1, fc1, 1);
  gemm_wmma_kernel<64><<<dim3(G / 16), dim3(32, 4), 0, stream>>>(fc1, fw2, fb2, out, 0);
}